// DirectedGraphLayer_6897717477506
// MI455X (gfx1250) — compile-verified
//
#include <hip/hip_runtime.h>

// Problem constants from the reference
#define BB   2
#define NN   50000
#define FIN  128
#define FOUT 64

typedef __attribute__((ext_vector_type(2))) float v2f;
typedef __attribute__((ext_vector_type(8))) float v8f;

// ---------------------------------------------------------------------------
// Kernel 1: fused dual GEMM via V_WMMA_F32_16X16X4_F32.
//   x:(B*N,128) row-major.  W,Wself:(128,64) row-major.
//   Writes xr[n*128 + b*64 + o]      = (x@W)[b,n,o]        (workspace)
//   Writes out[(b*N+n)*64 + o]       = (x@Wself)[b,n,o]+b_self[o]  (pre-ReLU)
// Block = 128 threads = 4 waves; wave w owns column tile [16w, 16w+16).
// Each block covers 16 consecutive flat rows (tile never straddles batches
// since N % 16 == 0).
// A-fragment layout (ISA 7.12.2, 32-bit A 16x4): lane r=lane&15 is row M=r,
// half=lane>>4 selects K pair {2h, 2h+1}  -> one contiguous b64 load.
// B-fragment (mirror): b.x = W[(k0+2h)][col], b.y = W[(k0+2h+1)][col].
// ---------------------------------------------------------------------------
__global__ void dgl_gemm_wmma(const float* __restrict__ x,
                              const float* __restrict__ W,
                              const float* __restrict__ Wself,
                              const float* __restrict__ bself,
                              float* __restrict__ xr,
                              float* __restrict__ out) {
  const int lane = threadIdx.x & 31;
  const int wave = threadIdx.x >> 5;      // 0..3 -> column tile
  const int half = lane >> 4;             // 0 or 1
  const int r    = lane & 15;             // row within tile / col within tile
  const long row0 = (long)blockIdx.x * 16;
  const int  col  = wave * 16 + r;        // 0..63

  v8f accW = {};   // x @ W
  v8f accS = {};   // x @ W_self

  const float* aptr = x + (row0 + r) * FIN + 2 * half;

  #pragma unroll 4
  for (int k0 = 0; k0 < FIN; k0 += 4) {
    // A fragment: one 64-bit load per lane (K = k0+2h, k0+2h+1)
    v2f a = *(const v2f*)(aptr + k0);
    const int kk = k0 + 2 * half;
    v2f bw; bw.x = W[kk * FOUT + col];      bw.y = W[(kk + 1) * FOUT + col];
    v2f bs; bs.x = Wself[kk * FOUT + col];  bs.y = Wself[(kk + 1) * FOUT + col];
    // Two independent accumulator chains -> hides WMMA->WMMA RAW hazard.
    accW = __builtin_amdgcn_wmma_f32_16x16x4_f32(
        false, a, false, bw, (short)0, accW, false, false);
    accS = __builtin_amdgcn_wmma_f32_16x16x4_f32(
        false, a, false, bs, (short)0, accS, false, false);
  }

  // C/D layout (ISA 7.12.2): VGPR i, lanes 0-15 -> M=i, lanes 16-31 -> M=8+i,
  // N = r.
  const int  bidx = (int)(row0 / NN);        // uniform within tile (N%16==0)
  const long n0   = row0 % NN;
  const float bias = bself[col];

  #pragma unroll
  for (int i = 0; i < 8; ++i) {
    const long nn = n0 + i + 8 * half;
    xr[nn * (BB * FOUT) + bidx * FOUT + col] = accW[i];
    out[((long)bidx * NN + nn) * FOUT + col] = accS[i] + bias;
  }
}

// ---------------------------------------------------------------------------
// Kernel 2: edge gather + scatter-add.  One wave per edge (grid-stride).
// Lane l handles 4 consecutive floats of the 128-wide xr row:
//   j = 4l..4l+3 ;  b = j>>6 (uniform per lane) ; o = j & 63.
// xr rows are L2-resident (25.6 MB); atomics land in L2 atomic units.
// ---------------------------------------------------------------------------
__global__ void dgl_edge_scatter(const int* __restrict__ erow,
                                 const int* __restrict__ ecol,
                                 const float* __restrict__ eval,
                                 const float* __restrict__ xr,
                                 float* __restrict__ out,
                                 int E) {
  const int lane    = threadIdx.x & 31;
  const int gwave   = (blockIdx.x * blockDim.x + threadIdx.x) >> 5;
  const int nwaves  = (gridDim.x * blockDim.x) >> 5;

  const int bidx = lane >> 4;             // batch index for this lane's slice
  const int o    = (lane * 4) & 63;       // column within FOUT

  for (int e = gwave; e < E; e += nwaves) {
    const int   rr = erow[e];
    const int   cc = ecol[e];
    const float v  = eval[e];
    const float4 xv = *(const float4*)(xr + (long)cc * (BB * FOUT) + lane * 4);
    float* dst = out + ((long)bidx * NN + rr) * FOUT + o;
    atomicAdd(dst + 0, v * xv.x);
    atomicAdd(dst + 1, v * xv.y);
    atomicAdd(dst + 2, v * xv.z);
    atomicAdd(dst + 3, v * xv.w);
  }
}

// ---------------------------------------------------------------------------
// Kernel 3: in-place ReLU over d_out (float4 vectorized; count % 4 == 0).
// ---------------------------------------------------------------------------
__global__ void dgl_relu(float* __restrict__ out, long n4) {
  long i = (long)blockIdx.x * blockDim.x + threadIdx.x;
  long stride = (long)gridDim.x * blockDim.x;
  for (; i < n4; i += stride) {
    float4 v = ((float4*)out)[i];
    v.x = v.x > 0.f ? v.x : 0.f;
    v.y = v.y > 0.f ? v.y : 0.f;
    v.z = v.z > 0.f ? v.z : 0.f;
    v.w = v.w > 0.f ? v.w : 0.f;
    ((float4*)out)[i] = v;
  }
}

extern "C" void kernel_launch(void* const* d_in, const int* in_sizes, int n_in,
                              void* d_out, int out_size, void* d_ws, size_t ws_size,
                              hipStream_t stream) {
  const float* x     = (const float*)d_in[0];
  const float* W     = (const float*)d_in[1];
  const float* Wself = (const float*)d_in[2];
  const float* bself = (const float*)d_in[3];
  const int*   erow  = (const int*)d_in[4];
  const int*   ecol  = (const int*)d_in[5];
  const float* eval  = (const float*)d_in[6];
  float*       out   = (float*)d_out;
  float*       xr    = (float*)d_ws;            // N * B*FOUT floats = 25.6 MB
  const int    E     = in_sizes[4];

  // 1) Fused dual WMMA GEMM: (B*N)/16 row tiles, 4 waves per block.
  dgl_gemm_wmma<<<(BB * NN) / 16, 128, 0, stream>>>(x, W, Wself, bself, xr, out);

  // 2) Edge scatter: 2048 blocks x 256 threads = 16384 waves, grid-stride.
  dgl_edge_scatter<<<2048, 256, 0, stream>>>(erow, ecol, eval, xr, out, E);

  // 3) ReLU in place.
  const long n4 = (long)out_size / 4;
  dgl_relu<<<1024, 256, 0, stream>>>(out, n4);
}